// BernNet_47364899340878
// MI455X (gfx1250) — compile-verified
//
#include <hip/hip_runtime.h>
#include <math.h>

typedef __attribute__((ext_vector_type(2))) float v2f;
typedef __attribute__((ext_vector_type(8))) float v8f;

#define F_IN 500
#define HID  64
#define NCLS 40

// ---------------- utility ----------------
__global__ void zero4_kernel(float4* __restrict__ p, int n4) {
    int i = blockIdx.x * blockDim.x + threadIdx.x;
    if (i < n4) p[i] = make_float4(0.f, 0.f, 0.f, 0.f);
}

__global__ void zero_kernel(float* __restrict__ p, int n) {
    int i = blockIdx.x * blockDim.x + threadIdx.x;
    if (i < n) p[i] = 0.0f;
}

// ---------------- degree / dis ----------------
__global__ void deg_kernel(const int* __restrict__ src, float* __restrict__ deg, int ne) {
    int e = blockIdx.x * blockDim.x + threadIdx.x;
    if (e < ne) atomicAdd(&deg[src[e]], 1.0f);
}

__global__ void dis_kernel(const float* __restrict__ deg, float* __restrict__ dis, int n) {
    int i = blockIdx.x * blockDim.x + threadIdx.x;
    if (i < n) {
        float d = deg[i];
        dis[i] = (d > 0.0f) ? (1.0f / sqrtf(d)) : 0.0f;
    }
}

// ---------------- GEMM1: h1 = relu(x @ W1 + b1), fp32 WMMA 16x16x4 ----------------
// block = 128 threads = 4 waves; wave w -> column tile w (16 cols), block -> 16 rows.
__global__ __launch_bounds__(128) void gemm1_relu_kernel(
    const float* __restrict__ x, const float* __restrict__ W1,
    const float* __restrict__ b1, float* __restrict__ h1, int n)
{
    const int lane = threadIdx.x & 31;
    const int wave = threadIdx.x >> 5;           // 0..3 -> col tile
    const int row0 = blockIdx.x * 16;
    const int col0 = wave * 16;
    const int lrow = lane & 15;
    const int lk   = (lane >> 4) * 2;            // 0 or 2 (K sub-offset)

    int arow = row0 + lrow; if (arow > n - 1) arow = n - 1;
    const float* __restrict__ xr = x + (long)arow * F_IN;
    const float* __restrict__ wc = W1 + col0 + lrow;   // lane's column of W1

    v8f c = {0.f,0.f,0.f,0.f,0.f,0.f,0.f,0.f};
    for (int k0 = 0; k0 < F_IN; k0 += 4) {
        v2f a, b;
        a.x = xr[k0 + lk];
        a.y = xr[k0 + lk + 1];
        b.x = wc[(k0 + lk)     * HID];
        b.y = wc[(k0 + lk + 1) * HID];
        c = __builtin_amdgcn_wmma_f32_16x16x4_f32(false, a, false, b,
                                                  (short)0, c, false, false);
    }
    const int col  = col0 + lrow;
    const float bias = b1[col];
    const int rsel = (lane >> 4) * 8;
#pragma unroll
    for (int v = 0; v < 8; ++v) {
        int row = row0 + v + rsel;
        if (row < n) {
            float val = c[v] + bias;
            h1[(long)row * HID + col] = val > 0.0f ? val : 0.0f;
        }
    }
}

// ---------------- GEMM2: t = h1 @ W2 + b2, fp32 WMMA, N=40 padded to 48 ----------------
// block = 96 threads = 3 waves; wave w -> col tile w (cols 0..47, masked to <40)
__global__ __launch_bounds__(96) void gemm2_kernel(
    const float* __restrict__ h1, const float* __restrict__ W2,
    const float* __restrict__ b2, float* __restrict__ t, int n)
{
    const int lane = threadIdx.x & 31;
    const int wave = threadIdx.x >> 5;           // 0..2
    const int row0 = blockIdx.x * 16;
    const int col0 = wave * 16;
    const int lrow = lane & 15;
    const int lk   = (lane >> 4) * 2;

    int arow = row0 + lrow; if (arow > n - 1) arow = n - 1;
    const float* __restrict__ hr = h1 + (long)arow * HID;
    const int col = col0 + lrow;
    const bool colok = (col < NCLS);

    v8f c = {0.f,0.f,0.f,0.f,0.f,0.f,0.f,0.f};
    for (int k0 = 0; k0 < HID; k0 += 4) {
        v2f a, b;
        a.x = hr[k0 + lk];
        a.y = hr[k0 + lk + 1];
        b.x = colok ? W2[(k0 + lk)     * NCLS + col] : 0.0f;
        b.y = colok ? W2[(k0 + lk + 1) * NCLS + col] : 0.0f;
        c = __builtin_amdgcn_wmma_f32_16x16x4_f32(false, a, false, b,
                                                  (short)0, c, false, false);
    }
    if (colok) {
        const float bias = b2[col];
        const int rsel = (lane >> 4) * 8;
#pragma unroll
        for (int v = 0; v < 8; ++v) {
            int row = row0 + v + rsel;
            if (row < n) t[row * NCLS + col] = c[v] + bias;
        }
    }
}

// ---------------- propagation (all float4: NCLS=40 is a multiple of 4) ----------------
// acc += cj * relu(temp[j]) * t    (vectorized; used once, for j=K)
__global__ void axpy4_kernel(float4* __restrict__ acc, const float4* __restrict__ t,
                             const float* __restrict__ temp, int j, float cj, int n4)
{
    int i = blockIdx.x * blockDim.x + threadIdx.x;
    if (i < n4) {
        float th = temp[j]; th = th > 0.0f ? th : 0.0f;
        const float k = cj * th;
        float4 a = acc[i], tv = t[i];
        a.x += k * tv.x; a.y += k * tv.y; a.z += k * tv.z; a.w += k * tv.w;
        acc[i] = a;
    }
}

// s[dst,f] += v[src,f] * dis[src]   (one thread per (edge, feature))
__global__ void scatter_kernel(const int* __restrict__ src, const int* __restrict__ dst,
                               const float* __restrict__ dis, const float* __restrict__ v,
                               float* __restrict__ s, int ne)
{
    int i = blockIdx.x * blockDim.x + threadIdx.x;
    int e = i / NCLS;
    if (e < ne) {
        int f  = i - e * NCLS;
        int se = src[e];
        int de = dst[e];
        atomicAdd(&s[de * NCLS + f], v[se * NCLS + f] * dis[se]);
    }
}

// acc -= dis[node]*s ; s = 0        (applies L in-place)
__global__ void applyL4_kernel(float4* __restrict__ acc, const float* __restrict__ dis,
                               float4* __restrict__ s, int n4)
{
    int i = blockIdx.x * blockDim.x + threadIdx.x;
    if (i < n4) {
        const float d = dis[i / (NCLS / 4)];          // 4 elems never cross a row
        float4 sv = s[i];
        s[i] = make_float4(0.f, 0.f, 0.f, 0.f);
        float4 a = acc[i];
        a.x -= d * sv.x; a.y -= d * sv.y; a.z -= d * sv.z; a.w -= d * sv.w;
        acc[i] = a;
    }
}

// t += dis[node]*s ; s = 0 ; optionally acc += cj*relu(temp[j])*t_new
// (fuses next iteration's axpy into the M application; acc already has L applied)
__global__ void applyM_axpy4_kernel(float4* __restrict__ t, const float* __restrict__ dis,
                                    float4* __restrict__ s, float4* __restrict__ acc,
                                    const float* __restrict__ temp, int j, float cj,
                                    int do_axpy, int n4)
{
    int i = blockIdx.x * blockDim.x + threadIdx.x;
    if (i < n4) {
        const float d = dis[i / (NCLS / 4)];
        float4 sv = s[i];
        s[i] = make_float4(0.f, 0.f, 0.f, 0.f);
        float4 tv = t[i];
        tv.x += d * sv.x; tv.y += d * sv.y; tv.z += d * sv.z; tv.w += d * sv.w;
        t[i] = tv;
        if (do_axpy) {
            float th = temp[j]; th = th > 0.0f ? th : 0.0f;
            const float k = cj * th;
            float4 a = acc[i];
            a.x += k * tv.x; a.y += k * tv.y; a.z += k * tv.z; a.w += k * tv.w;
            acc[i] = a;
        }
    }
}

// ---------------- out = c0*relu(temp[0])*t + acc ; log_softmax ----------------
__global__ void final_kernel(const float* __restrict__ t, const float* __restrict__ acc,
                             const float* __restrict__ temp, float c0,
                             float* __restrict__ out, int n)
{
    int row = blockIdx.x * blockDim.x + threadIdx.x;
    if (row >= n) return;
    float th0 = temp[0]; th0 = th0 > 0.0f ? th0 : 0.0f;
    const float k0 = c0 * th0;
    float o[NCLS];
    float m = -3.402823466e38f;
#pragma unroll
    for (int f = 0; f < NCLS; ++f) {
        float v = k0 * t[row * NCLS + f] + acc[row * NCLS + f];
        o[f] = v;
        m = v > m ? v : m;
    }
    float sum = 0.0f;
#pragma unroll
    for (int f = 0; f < NCLS; ++f) sum += expf(o[f] - m);
    float lse = logf(sum);
#pragma unroll
    for (int f = 0; f < NCLS; ++f) {
        out[row * NCLS + f] = o[f] - m - lse;            // log_softmax
        out[(long)n * NCLS + row * NCLS + f] = o[f];     // raw out
    }
}

// ---------------- host launcher ----------------
static inline size_t align16f(size_t nfloats) { return (nfloats + 3) & ~(size_t)3; }

extern "C" void kernel_launch(void* const* d_in, const int* in_sizes, int n_in,
                              void* d_out, int out_size, void* d_ws, size_t ws_size,
                              hipStream_t stream)
{
    const float* x    = (const float*)d_in[0];
    const int*   ei   = (const int*)  d_in[1];
    const float* W1   = (const float*)d_in[2];
    const float* b1   = (const float*)d_in[3];
    const float* W2   = (const float*)d_in[4];
    const float* b2   = (const float*)d_in[5];
    const float* temp = (const float*)d_in[6];
    float* out = (float*)d_out;

    const int n  = in_sizes[0] / F_IN;      // 100000
    const int ne = in_sizes[1] / 2;         // 3200000
    const int K  = in_sizes[6] - 1;         // 10
    const int* src = ei;
    const int* dst = ei + ne;

    // workspace layout (floats, 16B-aligned sub-buffers)
    float* ws  = (float*)d_ws;
    size_t off = 0;
    float* deg = ws + off; off += align16f(n);
    float* dis = ws + off; off += align16f(n);
    float* h1  = ws + off; off += align16f((size_t)n * HID);
    float* t   = ws + off; off += align16f((size_t)n * NCLS);
    float* acc = ws + off; off += align16f((size_t)n * NCLS);
    float* s   = ws + off;
    const int total = n * NCLS;
    const int n4 = total / 4;               // NCLS % 4 == 0

    // binomial coefficients c[j] = C(K,j)/2^K
    double cb[64];
    cb[0] = 1.0;
    for (int j = 1; j <= K; ++j) cb[j] = cb[j - 1] * (double)(K - j + 1) / (double)j;
    double scale = 1.0; for (int j = 0; j < K; ++j) scale *= 0.5;
    float c[64];
    for (int j = 0; j <= K; ++j) c[j] = (float)(cb[j] * scale);

    const int TB = 256;
    const int g4 = (n4 + TB - 1) / TB;
    // init: zero deg, acc, s
    zero_kernel<<<(n + TB - 1) / TB, TB, 0, stream>>>(deg, n);
    zero4_kernel<<<g4, TB, 0, stream>>>((float4*)acc, n4);
    zero4_kernel<<<g4, TB, 0, stream>>>((float4*)s, n4);

    // degrees & D^{-1/2}
    deg_kernel<<<(ne + TB - 1) / TB, TB, 0, stream>>>(src, deg, ne);
    dis_kernel<<<(n + TB - 1) / TB, TB, 0, stream>>>(deg, dis, n);

    // MLP encoder (WMMA fp32)
    const int rtiles = (n + 15) / 16;
    gemm1_relu_kernel<<<rtiles, 128, 0, stream>>>(x, W1, b1, h1, n);
    gemm2_kernel<<<rtiles, 96, 0, stream>>>(h1, W2, b2, t, n);

    // fused power + Horner sweep. Reference order, iteration i (j = K-i):
    //   acc = L(acc + c[j]*theta[j]*tmp[i]) ; tmp[i+1] = M(tmp[i])
    // The axpy for iteration i+1 is fused into the M application of iteration i.
    const int egrid = ((ne * NCLS) + TB - 1) / TB;
    axpy4_kernel<<<g4, TB, 0, stream>>>((float4*)acc, (const float4*)t, temp, K, c[K], n4);
    for (int i = 0; i < K; ++i) {
        const int j = K - i;
        scatter_kernel<<<egrid, TB, 0, stream>>>(src, dst, dis, acc, s, ne);
        applyL4_kernel<<<g4, TB, 0, stream>>>((float4*)acc, dis, (float4*)s, n4);
        scatter_kernel<<<egrid, TB, 0, stream>>>(src, dst, dis, t, s, ne);
        const int do_axpy = (i < K - 1) ? 1 : 0;
        applyM_axpy4_kernel<<<g4, TB, 0, stream>>>((float4*)t, dis, (float4*)s,
                                                   (float4*)acc, temp,
                                                   j - 1, (j >= 1) ? c[j - 1] : 0.0f,
                                                   do_axpy, n4);
    }

    // out = c0*theta0*t + acc ; log_softmax
    final_kernel<<<(n + TB - 1) / TB, TB, 0, stream>>>(t, acc, temp, c[0], out, n);
}